// UniformMPS_46686294507712
// MI455X (gfx1250) — compile-verified
//
#include <hip/hip_runtime.h>
#include <math.h>

typedef __attribute__((ext_vector_type(2))) float v2f;
typedef __attribute__((ext_vector_type(8))) float v8f;

// ---------------------------------------------------------------------------
// WMMA helpers: D(16x16,f32) = A(16x4,f32) * B(4x16,f32) + C, chained over 4
// K-slices for a full 16x16x16 f32 matmul on the matrix pipe.
// Layouts (wave32):
//   C/D element (row,col): vgpr = row&7, lane = col + 16*(row>=8)
//   A  element (M,K):      vgpr = K&1,  lane = M  + 16*(K>=2), slice q: K=4q+..
//   B  element (K,col):    vgpr = K&1,  lane = col+ 16*(K>=2), slice q
// ---------------------------------------------------------------------------
__device__ __forceinline__ v8f wmma4(v2f a, v2f b, v8f c) {
  return __builtin_amdgcn_wmma_f32_16x16x4_f32(
      /*neg_a=*/false, a, /*neg_b=*/false, b,
      /*c_mod=*/(short)0, c, /*reuse_a=*/false, /*reuse_b=*/false);
}

// Complex matmul (accumulating): (Dr + i*Di) += (Ar + i*Ai) @ (Br + i*Bi)
// bni must hold -Bi (f32 WMMA has no operand-negate modifier).
__device__ __forceinline__ void cwmma(const v2f* ar, const v2f* ai,
                                      const v2f* br, const v2f* bi,
                                      const v2f* bni, v8f& dr, v8f& di) {
#pragma unroll
  for (int q = 0; q < 4; ++q) dr = wmma4(ar[q], br[q], dr);
#pragma unroll
  for (int q = 0; q < 4; ++q) dr = wmma4(ai[q], bni[q], dr);
#pragma unroll
  for (int q = 0; q < 4; ++q) di = wmma4(ar[q], bi[q], di);
#pragma unroll
  for (int q = 0; q < 4; ++q) di = wmma4(ai[q], br[q], di);
}

// Read a 16x16 matrix from LDS into A-layout K-slices.
__device__ __forceinline__ void loadA(const float* lds, int lo, int hi, v2f* a) {
#pragma unroll
  for (int q = 0; q < 4; ++q) {
    int k = 4 * q + 2 * hi;
    a[q][0] = lds[lo * 16 + k];       // (M=lo, K=k)
    a[q][1] = lds[lo * 16 + k + 1];   // contiguous pair -> ds_load_b64
  }
}

// Read the TRANSPOSE of the LDS-resident matrix into A-layout slices.
__device__ __forceinline__ void loadAT(const float* lds, int lo, int hi, v2f* a) {
#pragma unroll
  for (int q = 0; q < 4; ++q) {
    int k = 4 * q + 2 * hi;
    a[q][0] = lds[k * 16 + lo];
    a[q][1] = lds[(k + 1) * 16 + lo];
  }
}

// Store C/D-layout accumulator to LDS (row-major), plain and transposed.
__device__ __forceinline__ void storeC(float* lds, int lo, int hi, v8f v) {
#pragma unroll
  for (int i = 0; i < 8; ++i) lds[(i + 8 * hi) * 16 + lo] = v[i];
}
__device__ __forceinline__ void storeCT(float* lds, int lo, int hi, v8f v) {
#pragma unroll
  for (int i = 0; i < 8; ++i) lds[lo * 16 + (i + 8 * hi)] = v[i];
}

// ---------------------------------------------------------------------------
// Kernel 1: per-sample transfer-matrix chain, 16 samples per wave (1 wave/block).
// State V: 16 samples x D=16 complex, kept in WMMA C-layout registers.
// Each step computes V@bulk[0] and V@bulk[1] with WMMA, selects per-row by spin.
// ---------------------------------------------------------------------------
__global__ __launch_bounds__(32) void mps_chain(
    const int* __restrict__ spins, const float* __restrict__ lr,
    const float* __restrict__ li, const float* __restrict__ br_g,
    const float* __restrict__ bi_g, const float* __restrict__ rr_g,
    const float* __restrict__ ri_g, float* __restrict__ partial, int L) {
  __shared__ float sr[256], si[256];
  __shared__ unsigned smask[64];
  __shared__ float samp[32];

  const int lane = threadIdx.x;
  const int lo = lane & 15, hi = lane >> 4;
  const int base = blockIdx.x * 16;

  // Per-step spin bitmasks (bit r = spin of sample r at step t), d==2.
  for (int t = lane; t < L; t += 32) {
    unsigned m = 0;
#pragma unroll
    for (int r = 0; r < 16; ++r)
      m |= (unsigned)(spins[(base + r) * L + t] != 0) << r;
    smask[t] = m;
  }

  // bulk matrices in B-layout registers: real, imag, -imag for both spins.
  v2f Br[2][4], Bi[2][4], Bni[2][4];
#pragma unroll
  for (int s = 0; s < 2; ++s)
#pragma unroll
    for (int q = 0; q < 4; ++q)
#pragma unroll
      for (int j = 0; j < 2; ++j) {
        int K = 4 * q + 2 * hi + j;
        float re = br_g[s * 256 + K * 16 + lo];
        float im = bi_g[s * 256 + K * 16 + lo];
        Br[s][q][j] = re; Bi[s][q][j] = im; Bni[s][q][j] = -im;
      }
  __syncthreads();

  // Init V from left[spin_0]: C-layout, row = sample, col = D index.
  v8f Vr, Vi;
  unsigned m0 = smask[0];
#pragma unroll
  for (int i = 0; i < 8; ++i) {
    int row = i + 8 * hi;
    unsigned s0 = (m0 >> row) & 1u;
    Vr[i] = lr[s0 * 16 + lo];
    Vi[i] = li[s0 * 16 + lo];
  }

  for (int t = 1; t <= L - 2; ++t) {
    __syncthreads();
    storeC(sr, lo, hi, Vr);
    storeC(si, lo, hi, Vi);
    __syncthreads();
    v2f Ar[4], Ai[4];
    loadA(sr, lo, hi, Ar);
    loadA(si, lo, hi, Ai);
    v8f d0r = {}, d0i = {}, d1r = {}, d1i = {};
    cwmma(Ar, Ai, Br[0], Bi[0], Bni[0], d0r, d0i);  // V @ bulk[0]
    cwmma(Ar, Ai, Br[1], Bi[1], Bni[1], d1r, d1i);  // V @ bulk[1]
    unsigned m = smask[t];
#pragma unroll
    for (int i = 0; i < 8; ++i) {
      int row = i + 8 * hi;
      bool take = (m >> row) & 1u;
      Vr[i] = take ? d1r[i] : d0r[i];
      Vi[i] = take ? d1i[i] : d0i[i];
    }
  }

  __syncthreads();
  storeC(sr, lo, hi, Vr);
  storeC(si, lo, hi, Vi);
  __syncthreads();

  // Amplitude a_n = v . right[spin_{L-1}]; per-sample log-prob on lanes 0..15.
  float logp = 0.f;
  if (hi == 0) {
    int slast = spins[(base + lo) * L + (L - 1)];
    float ar = 0.f, ai = 0.f;
#pragma unroll
    for (int c = 0; c < 16; ++c) {
      float vr = sr[lo * 16 + c], vi = si[lo * 16 + c];
      float rr = rr_g[slast * 16 + c], ri = ri_g[slast * 16 + c];
      ar += vr * rr - vi * ri;
      ai += vr * ri + vi * rr;
    }
    logp = logf(ar * ar + ai * ai);
  }
  samp[lane] = logp;
  __syncthreads();
  if (lane == 0) {
    float s = 0.f;
#pragma unroll
    for (int r = 0; r < 16; ++r) s += samp[r];
    partial[blockIdx.x] = s;  // deterministic per-block slot (no float atomics)
  }
}

// ---------------------------------------------------------------------------
// Kernel 2: norm iteration cont <- sum_s bulk[s]^T @ cont @ conj(bulk[s]),
// 62 times, single wave, same WMMA machinery; then combines with the chain
// partials to produce nll = -mean(log p) + log(norm).
// ---------------------------------------------------------------------------
__global__ __launch_bounds__(32) void mps_norm(
    const float* __restrict__ lr, const float* __restrict__ li,
    const float* __restrict__ br_g, const float* __restrict__ bi_g,
    const float* __restrict__ rr_g, const float* __restrict__ ri_g,
    const float* __restrict__ partial, int nblocks, int L, int N,
    float* __restrict__ out) {
  __shared__ float cr[256], ci[256], yr[256], yi[256];
  __shared__ float red[32];
  const int lane = threadIdx.x;
  const int lo = lane & 15, hi = lane >> 4;

  v2f Br[2][4], Bi[2][4], Bni[2][4];
#pragma unroll
  for (int s = 0; s < 2; ++s)
#pragma unroll
    for (int q = 0; q < 4; ++q)
#pragma unroll
      for (int j = 0; j < 2; ++j) {
        int K = 4 * q + 2 * hi + j;
        float re = br_g[s * 256 + K * 16 + lo];
        float im = bi_g[s * 256 + K * 16 + lo];
        Br[s][q][j] = re; Bi[s][q][j] = im; Bni[s][q][j] = -im;
      }

  // cont[j,k] = sum_s left[s,j] * conj(left[s,k])
#pragma unroll
  for (int i8 = 0; i8 < 8; ++i8) {
    int e = lane + 32 * i8;
    int j = e >> 4, k = e & 15;
    float creal = 0.f, cimag = 0.f;
#pragma unroll
    for (int s = 0; s < 2; ++s) {
      float a = lr[s * 16 + j], b = li[s * 16 + j];
      float c = lr[s * 16 + k], d = li[s * 16 + k];
      creal += a * c + b * d;
      cimag += b * c - a * d;
    }
    cr[e] = creal;
    ci[e] = cimag;
  }

  for (int it = 0; it < L - 2; ++it) {
    __syncthreads();
    v2f Ar[4], Ai[4];
    loadA(cr, lo, hi, Ar);
    loadA(ci, lo, hi, Ai);
    v8f aTr = {}, aTi = {};
#pragma unroll
    for (int s = 0; s < 2; ++s) {
      // Y = cont @ conj(bulk[s]): B = (Br, -Bi), so pass (Br, Bni, Bi).
      v8f Yr = {}, Yi = {};
      cwmma(Ar, Ai, Br[s], Bni[s], Bi[s], Yr, Yi);
      __syncthreads();
      storeC(yr, lo, hi, Yr);
      storeC(yi, lo, hi, Yi);
      __syncthreads();
      v2f Atr[4], Ati[4];
      loadAT(yr, lo, hi, Atr);  // A = Y^T
      loadAT(yi, lo, hi, Ati);
      // cont'^T += Y^T @ bulk[s]   (== (bulk[s]^T @ Y)^T)
      cwmma(Atr, Ati, Br[s], Bi[s], Bni[s], aTr, aTi);
    }
    __syncthreads();
    storeCT(cr, lo, hi, aTr);  // write back un-transposed cont'
    storeCT(ci, lo, hi, aTi);
  }
  __syncthreads();

  // norm = Re sum_{i,j} cont[i,j] * rc[i,j],  rc[i,j] = sum_s r[s,i]*conj(r[s,j])
  float part = 0.f;
#pragma unroll
  for (int i8 = 0; i8 < 8; ++i8) {
    int e = lane + 32 * i8;
    int i = e >> 4, j = e & 15;
    float rcr = 0.f, rci = 0.f;
#pragma unroll
    for (int s = 0; s < 2; ++s) {
      float a = rr_g[s * 16 + i], b = ri_g[s * 16 + i];
      float c = rr_g[s * 16 + j], d = ri_g[s * 16 + j];
      rcr += a * c + b * d;
      rci += b * c - a * d;
    }
    part += cr[e] * rcr - ci[e] * rci;
  }
  red[lane] = part;
  __syncthreads();
  if (lane == 0) {
    float norm = 0.f;
#pragma unroll
    for (int r = 0; r < 32; ++r) norm += red[r];
    float ssum = 0.f;
    for (int b = 0; b < nblocks; ++b) ssum += partial[b];  // fixed order
    out[0] = -(ssum / (float)N) + logf(norm);
  }
}

// ---------------------------------------------------------------------------
extern "C" void kernel_launch(void* const* d_in, const int* in_sizes, int n_in,
                              void* d_out, int out_size, void* d_ws,
                              size_t ws_size, hipStream_t stream) {
  const int* spins = (const int*)d_in[0];
  const float* lr = (const float*)d_in[1];
  const float* li = (const float*)d_in[2];
  const float* br = (const float*)d_in[3];
  const float* bi = (const float*)d_in[4];
  const float* rr = (const float*)d_in[5];
  const float* ri = (const float*)d_in[6];
  float* out = (float*)d_out;
  float* partial = (float*)d_ws;

  const int L = 64;                   // reference: spin_config shape (N, 64)
  const int N = in_sizes[0] / L;      // 8192
  const int nblocks = N / 16;         // 16 samples per wave32 block

  mps_chain<<<nblocks, 32, 0, stream>>>(spins, lr, li, br, bi, rr, ri, partial, L);
  mps_norm<<<1, 32, 0, stream>>>(lr, li, br, bi, rr, ri, partial, nblocks, L, N, out);
}